// my_model_23613730193919
// MI455X (gfx1250) — compile-verified
//
#include <hip/hip_runtime.h>
#include <stdint.h>

// Problem constants (from reference)
#define BB    64
#define CC    12
#define NCC   16
#define NN0   2048
#define NN1   1000
#define NN2   500
#define NN3   100
#define NDIMM 3

// Padded workspace column counts (multiples of 64: one wave strip = 64 cols)
#define N1P   1024
#define N2P   512
#define NYP   64
#define NR    1600   // 16 experts * 100 cols per class (25*64)

// Async-staged B tile geometry
#define KC        32                    // K floats per staged chunk (128B per row)
#define SMSTRIDE  36                    // padded LDS row stride in floats (144B, 16B aligned)
#define SMFLOATS  (64 * SMSTRIDE)       // floats per buffer (64 rows)

typedef __attribute__((ext_vector_type(2))) float v2f;
typedef __attribute__((ext_vector_type(8))) float v8f;

#define WMMA_F32(A, Bf, Cacc) \
  __builtin_amdgcn_wmma_f32_16x16x4_f32(false, (A), false, (Bf), (short)0, (Cacc), false, false)

// CDNA5 async tensor path: coalesced global -> LDS, tracked by ASYNCcnt.
__device__ __forceinline__ void async_b128(uint32_t lds_byte, const float* gaddr) {
  asm volatile("global_load_async_to_lds_b128 %0, %1, off"
               :: "v"(lds_byte), "v"(gaddr) : "memory");
}
__device__ __forceinline__ void wait_async_le16() {
  asm volatile("s_wait_asynccnt 0x10" ::: "memory");
}
__device__ __forceinline__ void wait_async_0() {
  asm volatile("s_wait_asynccnt 0x0" ::: "memory");
}

// ---------------------------------------------------------------------------
// Kernel 0: per-class compacted batch lists (deterministic sequential order).
// ---------------------------------------------------------------------------
__global__ void build_lists_kernel(const int* __restrict__ cl,
                                   int* __restrict__ counts,
                                   int* __restrict__ lists,
                                   int* __restrict__ slots) {
  if (threadIdx.x == 0 && blockIdx.x == 0) {
    for (int c = 0; c < CC; ++c) counts[c] = 0;
    for (int b = 0; b < BB; ++b) {
      int c = cl[b];
      if (c < 0) c = 0;
      if (c >= CC) c = CC - 1;
      int s = counts[c];
      counts[c] = s + 1;
      lists[c * BB + s] = b;
      slots[b] = s;
    }
  }
}

// ---------------------------------------------------------------------------
// Per-class GEMM: one wave computes a 16(M) x 64(N) strip with 4 f32-WMMA
// accumulators. B (the 280MB weight stream) is double-buffer staged into LDS
// with global_load_async_to_lds_b128; fragments come from LDS conflict-free.
// Steady-state loop is straight-line: issue(next) / wait<=16 / consume(cur).
// ---------------------------------------------------------------------------
__global__ void __launch_bounds__(32)
gemm_wmma_kernel(const float* __restrict__ X,      // gather source (gatherA=1)
                 const float* __restrict__ Abase,  // workspace A (gatherA=0)
                 const float* __restrict__ Bw,     // weights, row n has ldb cols
                 const float* __restrict__ bias,   // bias, [c*Nvalid + n]
                 float* __restrict__ Out,          // [c][64][ldo]
                 const int* __restrict__ counts,
                 const int* __restrict__ lists,
                 int lda, int ldb, int ldo,
                 int Nvalid, int K, int kvalid,
                 int gatherA, int do_relu) {
  __shared__ float smB[2][SMFLOATS];

  const int ns = blockIdx.x;       // 64-column N strip
  const int mt = blockIdx.y;       // 16-row M tile
  const int c  = blockIdx.z;       // class
  const int cnt = counts[c];
  if (mt * 16 >= cnt) return;      // uniform early-exit

  const int lane = threadIdx.x;    // wave32
  const int lo   = lane & 15;
  const int khi  = (lane >> 4) << 1;   // {0,2}: f32 A/B fragment K sub-offset
  const int lsub = lane & 7;           // 16B segment within a 128B row chunk
  const int rsub = lane >> 3;          // row 0..3 within a 4-row async group

  // A fragment row (m = lo)
  const int slot = mt * 16 + lo;
  const float* arow;
  if (gatherA) {
    const int bidx = (slot < cnt) ? lists[c * BB + slot] : 0;
    arow = X + (size_t)bidx * lda;
  } else {
    arow = Abase + (size_t)(c * BB + slot) * lda;
  }
  const float* ap = arow + khi;

  const float* bw_c = Bw + (size_t)c * Nvalid * ldb;
  const bool strip_clamped = (ns * 64 + 63) >= Nvalid;
  const float* gp0 = bw_c + (size_t)(ns * 64 + rsub) * ldb + lsub * 4;  // fast path base

  const uint32_t lds0 = (uint32_t)(uintptr_t)&smB[0][0];
  const uint32_t lbase_lane = lds0 + (uint32_t)(rsub * (SMSTRIDE * 4) + lsub * 16);

  // Issue one 64-row x 32-float chunk (16 x b128) into buffer bsel.
  auto issue = [&](int k0c, int bsel) {
    uint32_t l = lbase_lane + (uint32_t)bsel * (SMFLOATS * 4);
    const bool kcl = (k0c + KC) > kvalid;
    if (!strip_clamped && !kcl) {
      const float* gp = gp0 + k0c;
#pragma unroll
      for (int i = 0; i < 16; ++i) {
        async_b128(l, gp);
        gp += 4 * (size_t)ldb;
        l += SMSTRIDE * 4 * 4;
      }
    } else {  // boundary strip/chunk: clamp rows and 16B k-segments in-bounds
#pragma unroll
      for (int i = 0; i < 16; ++i) {
        int n = ns * 64 + 4 * i + rsub;
        if (n >= Nvalid) n = Nvalid - 1;
        int kc = k0c + lsub * 4;
        if (kc > ldb - 4) kc = ldb - 4;
        async_b128(l, bw_c + (size_t)n * ldb + kc);
        l += SMSTRIDE * 4 * 4;
      }
    }
  };

  v8f acc0 = {0.f,0.f,0.f,0.f,0.f,0.f,0.f,0.f};
  v8f acc1 = acc0, acc2 = acc0, acc3 = acc0;

  // Consume one staged chunk; guard==true zeroes B frags at k >= kvalid
  // (only possible on the final chunk since K - kvalid < KC here).
  auto consume = [&](int k0c, int bsel, bool guard) {
    const float* apc = ap + k0c;
    const float* sb = &smB[bsel][lo * SMSTRIDE + khi];
#pragma unroll
    for (int u = 0; u < 8; ++u) {
      v2f a  = *(const v2f*)(apc + u * 4);
      v2f b0 = *(const v2f*)(sb + u * 4);
      v2f b1 = *(const v2f*)(sb + u * 4 + 16 * SMSTRIDE);
      v2f b2 = *(const v2f*)(sb + u * 4 + 32 * SMSTRIDE);
      v2f b3 = *(const v2f*)(sb + u * 4 + 48 * SMSTRIDE);
      if (guard) {
        const v2f bz = {0.f, 0.f};
        const int k = k0c + u * 4 + khi;
        if (k >= kvalid) { b0 = bz; b1 = bz; b2 = bz; b3 = bz; }
      }
      acc0 = WMMA_F32(a, b0, acc0);
      acc1 = WMMA_F32(a, b1, acc1);
      acc2 = WMMA_F32(a, b2, acc2);
      acc3 = WMMA_F32(a, b3, acc3);
    }
  };

  const int nchunks = K / KC;       // K is a multiple of KC in all launches
  issue(0, 0);
  int k0c = 0, knext = KC;
  for (int ci = 0; ci < nchunks - 1; ++ci) {   // steady state: straight-line
    issue(knext, (ci + 1) & 1);
    wait_async_le16();
    consume(k0c, ci & 1, false);
    k0c = knext;
    knext += KC;
  }
  wait_async_0();                              // peeled last chunk (guarded)
  consume(k0c, (nchunks - 1) & 1, (k0c + KC) > kvalid);

  // Epilogue: bias (clamped index) + optional relu; pad cols carry finite
  // garbage that downstream stages multiply by zero-guarded B fragments.
  const int n0 = ns * 64 + lo;
  const int n1 = n0 + 16, n2 = n0 + 32, n3 = n0 + 48;
  const int mb = (lane >> 4) * 8;  // C/D layout: element v -> M = v + 8*(lane/16)
  float* outc = Out + (size_t)c * BB * ldo + (size_t)(mt * 16 + mb) * ldo;
  const float* bias_c = bias + (size_t)c * Nvalid;
  const float bv0 = bias_c[(n0 < Nvalid) ? n0 : Nvalid - 1];
  const float bv1 = bias_c[(n1 < Nvalid) ? n1 : Nvalid - 1];
  const float bv2 = bias_c[(n2 < Nvalid) ? n2 : Nvalid - 1];
  const float bv3 = bias_c[(n3 < Nvalid) ? n3 : Nvalid - 1];
#pragma unroll
  for (int v = 0; v < 8; ++v) {
    float r0 = acc0[v] + bv0, r1 = acc1[v] + bv1;
    float r2 = acc2[v] + bv2, r3 = acc3[v] + bv3;
    if (do_relu) {
      r0 = fmaxf(r0, 0.f); r1 = fmaxf(r1, 0.f);
      r2 = fmaxf(r2, 0.f); r3 = fmaxf(r3, 0.f);
    }
    float* orow = outc + (size_t)v * ldo;
    orow[n0] = r0; orow[n1] = r1; orow[n2] = r2; orow[n3] = r3;
  }
}

// ---------------------------------------------------------------------------
// Finalize: copy y1 row, first-max argmax (matches jnp.argmax), then the tiny
// length-100 refine dots for the selected expert only.
// ---------------------------------------------------------------------------
__global__ void __launch_bounds__(32)
finalize_kernel(const int* __restrict__ cl, const int* __restrict__ slots,
                const float* __restrict__ Y1g, const float* __restrict__ Rg,
                const float* __restrict__ W2r, const float* __restrict__ b2r,
                float* __restrict__ out) {
  const int b = blockIdx.x;
  int c = cl[b];
  if (c < 0) c = 0;
  if (c >= CC) c = CC - 1;
  const int s = slots[b];
  const float* y1row = Y1g + (size_t)(c * BB + s) * NYP;
  const int t = threadIdx.x;

  if (t < NCC) out[b * NCC + t] = y1row[t];

  int pose = 0;
  float best = y1row[0];
  for (int j = 1; j < NCC; ++j) {
    float v = y1row[j];
    if (v > best) { best = v; pose = j; }
  }
  const int e = c * NCC + pose;
  const float* rrow = Rg + (size_t)(c * BB + s) * NR + pose * NN3;
  if (t < NDIMM) {
    const float* w = W2r + (size_t)(e * NDIMM + t) * NN3;
    float a = b2r[e * NDIMM + t];
    for (int j = 0; j < NN3; ++j) a += rrow[j] * w[j];
    out[BB * NCC + b * NDIMM + t] = a;
  }
}

// ---------------------------------------------------------------------------
extern "C" void kernel_launch(void* const* d_in, const int* in_sizes, int n_in,
                              void* d_out, int out_size, void* d_ws, size_t ws_size,
                              hipStream_t stream) {
  (void)in_sizes; (void)n_in; (void)out_size; (void)ws_size;

  const float* x   = (const float*)d_in[0];
  const int*   cl  = (const int*)  d_in[1];
  const float* W1b = (const float*)d_in[2];
  const float* b1b = (const float*)d_in[3];
  const float* W2b = (const float*)d_in[4];
  const float* b2b = (const float*)d_in[5];
  const float* W3b = (const float*)d_in[6];
  const float* b3b = (const float*)d_in[7];
  const float* W1r = (const float*)d_in[8];
  const float* b1r = (const float*)d_in[9];
  const float* W2r = (const float*)d_in[10];
  const float* b2r = (const float*)d_in[11];
  float* out = (float*)d_out;

  // Workspace layout (~9.9 MB)
  char* ws = (char*)d_ws;
  int* counts = (int*)(ws + 0);      // 12
  int* lists  = (int*)(ws + 256);    // 12*64
  int* slots  = (int*)(ws + 3584);   // 64
  float* H1  = (float*)(ws + 4096);            // [C][64][N1P]
  float* H2  = H1  + (size_t)CC * BB * N1P;    // [C][64][N2P]
  float* Y1g = H2  + (size_t)CC * BB * N2P;    // [C][64][NYP]
  float* Rg  = Y1g + (size_t)CC * BB * NYP;    // [C][64][NR]

  build_lists_kernel<<<1, 32, 0, stream>>>(cl, counts, lists, slots);

  // G1: H1 = relu(x_g @ W1b[c]^T + b1b[c])   N=1000 (pad 1024), K=2048
  gemm_wmma_kernel<<<dim3(N1P / 64, 4, CC), 32, 0, stream>>>(
      x, nullptr, W1b, b1b, H1, counts, lists,
      NN0, NN0, N1P, NN1, NN0, NN0, /*gatherA=*/1, /*relu=*/1);

  // G2: H2 = relu(H1 @ W2b[c]^T + b2b[c])    N=500 (pad 512), K=1024 (kvalid 1000)
  gemm_wmma_kernel<<<dim3(N2P / 64, 4, CC), 32, 0, stream>>>(
      nullptr, H1, W2b, b2b, H2, counts, lists,
      N1P, NN1, N2P, NN2, N1P, NN1, /*gatherA=*/0, /*relu=*/1);

  // G3: Y1g = H2 @ W3b[c]^T + b3b[c]         N=16 (pad 64), K=512 (kvalid 500)
  gemm_wmma_kernel<<<dim3(NYP / 64, 4, CC), 32, 0, stream>>>(
      nullptr, H2, W3b, b3b, Y1g, counts, lists,
      N2P, NN2, NYP, NCC, N2P, NN2, /*gatherA=*/0, /*relu=*/0);

  // GR1: Rg = relu(x_g @ W1r[16 experts of c]^T + b1r)  N=1600, K=2048
  gemm_wmma_kernel<<<dim3(NR / 64, 4, CC), 32, 0, stream>>>(
      x, nullptr, W1r, b1r, Rg, counts, lists,
      NN0, NN0, NR, NR, NN0, NN0, /*gatherA=*/1, /*relu=*/1);

  finalize_kernel<<<BB, 32, 0, stream>>>(cl, slots, Y1g, Rg, W2r, b2r, out);
}